// PrecondTiming_50474455663126
// MI455X (gfx1250) — compile-verified
//
#include <hip/hip_runtime.h>

// Scatter-add for timing-net preconditioner:
//   out[node] += beta * w[tnet]  for both pins of every tnet arc.
//
// Inputs (setup_inputs order):
//   d_in[0] beta          float32 [1]
//   d_in[1] tnet_weights  float32 [NUM_TNETS]
//   d_in[2] flat_tnet2pin int64   [2*NUM_TNETS]   (values < 2^25, low dword suffices)
//   d_in[3] pin2node_map  int64   [NUM_PINS]      (values < 2^20, low dword suffices)
// Output: float32 [NUM_NODES]
//
// Roofline: ~324 MiB mandatory traffic -> ~14 us at 23.3 TB/s HBM.
// pin2node (160 MiB) is kept L2-resident (192 MB L2); the streaming index and
// weight loads are non-temporal so they don't evict it. The 16.7M f32 adds are
// non-returning relaxed agent-scope atomics resolved by the L2 atomic units
// over a 4 MiB accumulator. Gather/atomic indices are unsigned 32-bit so the
// backend can use SGPR-base + u32-VGPR-offset (GVS/saddr) addressing with no
// per-element 64-bit address arithmetic.

typedef int v4i __attribute__((ext_vector_type(4)));

#define BS 256   // threads per block
#define T  4     // tnets per thread

// ---------------------------------------------------------------------------
// Gather + scatter-add. Block owns T*BS consecutive tnets; iteration i reads
// tnet (tile + i*BS + tid) so every load instruction is a contiguous
// 512B-per-wave (b128) / 128B-per-wave (b32) transaction.
//   phase 1: stream indices (b128 NT) + weights (b32 NT), pre-scale by beta
//   phase 2: 8 independent gathers from pin2node (low dword, L2-resident)
//   phase 3: 8 non-returning global_atomic_add_f32 into the 4 MiB accumulator
// ---------------------------------------------------------------------------
__global__ __launch_bounds__(BS) void pt_scatter(
    const float* __restrict__ beta,
    const float* __restrict__ weights,
    const int*  __restrict__ pins,      // flat_tnet2pin as i32; tnet t -> ints [4t..4t+3]
    const int*  __restrict__ pin2node,  // pin2node_map as i32; node = pin2node[2*pin]
    float* __restrict__ out,
    int num_tnets)
{
    const float b = beta[0];  // uniform -> s_load + broadcast
    const long tile = (long)blockIdx.x * (T * BS);

    unsigned p0[T], p1[T];
    float    wv[T];
    unsigned n0[T], n1[T];

    if (tile + T * BS <= (long)num_tnets) {
        // -------- full tile: unguarded fast path --------
        #pragma unroll
        for (int i = 0; i < T; ++i) {
            long ti = tile + i * BS + threadIdx.x;
            v4i v = __builtin_nontemporal_load((const v4i*)(pins + 4 * ti)); // 16B aligned
            p0[i] = (unsigned)v.x;
            p1[i] = (unsigned)v.z;
            wv[i] = __builtin_nontemporal_load(weights + ti) * b;
        }
        #pragma unroll
        for (int i = 0; i < T; ++i) {
            // u32 index -> zero-extend -> saddr-form global_load_b32
            n0[i] = (unsigned)pin2node[2u * p0[i]];
            n1[i] = (unsigned)pin2node[2u * p1[i]];
        }
        #pragma unroll
        for (int i = 0; i < T; ++i) {
            __hip_atomic_fetch_add(out + n0[i], wv[i],
                                   __ATOMIC_RELAXED, __HIP_MEMORY_SCOPE_AGENT);
            __hip_atomic_fetch_add(out + n1[i], wv[i],
                                   __ATOMIC_RELAXED, __HIP_MEMORY_SCOPE_AGENT);
        }
    } else {
        // -------- partial tile: guarded tail path --------
        #pragma unroll
        for (int i = 0; i < T; ++i) {
            long ti = tile + i * BS + threadIdx.x;
            if (ti < (long)num_tnets) {
                v4i v = __builtin_nontemporal_load((const v4i*)(pins + 4 * ti));
                unsigned node0 = (unsigned)pin2node[2u * (unsigned)v.x];
                unsigned node1 = (unsigned)pin2node[2u * (unsigned)v.z];
                float w = __builtin_nontemporal_load(weights + ti) * b;
                __hip_atomic_fetch_add(out + node0, w,
                                       __ATOMIC_RELAXED, __HIP_MEMORY_SCOPE_AGENT);
                __hip_atomic_fetch_add(out + node1, w,
                                       __ATOMIC_RELAXED, __HIP_MEMORY_SCOPE_AGENT);
            }
        }
    }
}

extern "C" void kernel_launch(void* const* d_in, const int* in_sizes, int n_in,
                              void* d_out, int out_size, void* d_ws, size_t ws_size,
                              hipStream_t stream) {
    const float* beta     = (const float*)d_in[0];
    const float* weights  = (const float*)d_in[1];
    const int*   pins     = (const int*)d_in[2];   // int64 data, low-dword access
    const int*   pin2node = (const int*)d_in[3];   // int64 data, low-dword access
    float*       out      = (float*)d_out;

    const int num_tnets = in_sizes[1];

    // Zero the accumulator via the driver fill path (graph-capturable).
    hipMemsetAsync(out, 0, (size_t)out_size * sizeof(float), stream);

    // Scatter-add: each block covers T*BS tnets.
    long tiles = ((long)num_tnets + (T * BS) - 1) / (T * BS);
    pt_scatter<<<(int)tiles, BS, 0, stream>>>(beta, weights, pins, pin2node,
                                              out, num_tnets);
}